// lossFunc_32435593019868
// MI455X (gfx1250) — compile-verified
//
#include <hip/hip_runtime.h>
#include <stdint.h>

// Problem constants (from the reference): B=1024, S=200, Q=512
#define NB     1024
#define NS     200
#define SM1    199      // S-1 pairs per student
#define NQ     512
#define ROW    1024     // batch row = 2*Q floats (4 KB)

#define BDIM   256      // 8 waves (wave32)
#define WAVES  8
#define KP     8        // pairs per wave (software pipeline depth over rows)
// pairs per block = WAVES*KP = 64 ; total pairs = 1024*199 = 203776 = 64 * 3184

typedef __attribute__((address_space(3))) float lds_float;

// Low 32 bits of a generic LDS pointer = wave-relative LDS byte offset
__device__ __forceinline__ unsigned lds_off(const float* p) {
  return (unsigned)(uintptr_t)(lds_float*)p;
}

// CDNA5 async DMA: copy 16B/lane global->LDS, tracked by ASYNCcnt.
// TH_LOAD_NT: the 839 MB batch stream is single-use; don't thrash the 192 MB L2
// (keep it for the sparse pred gather lines instead).
__device__ __forceinline__ void async_b128_nt(const float* g, unsigned loff) {
  asm volatile("global_load_async_to_lds_b128 %0, %1, off th:TH_LOAD_NT"
               :: "v"(loff), "v"((uint64_t)(uintptr_t)g)
               : "memory");
}

// Copy one 4 KB batch row (1024 floats) into LDS: 8 async b128 ops (512 B each).
__device__ __forceinline__ void issue_row(const float* rowBase, const float* ldsBase, int lane) {
  const float* g   = rowBase + lane * 4;
  unsigned     off = lds_off(ldsBase) + lane * 16;
#pragma unroll
  for (int j = 0; j < 8; ++j)
    async_b128_nt(g + j * 128, off + j * 512);
}

// advance (b,s) to the next pair without a division
__device__ __forceinline__ void advance(int& b, int& s) {
  if (++s == SM1) { s = 0; ++b; }
}

__global__ void __launch_bounds__(BDIM) dkt_loss_stage1(
    const float* __restrict__ pred,    // [B, S, Q]
    const float* __restrict__ batch,   // [B, S, 2Q]
    float* __restrict__ partials) {
  // 8 waves * 2 buffers * 4 KB = 64 KB LDS (double-buffered per wave)
  __shared__ __align__(16) float smem[WAVES * 2 * ROW];

  const int tid  = threadIdx.x;
  const int lane = tid & 31;
  const int wave = tid >> 5;

  float* buf0 = &smem[wave * 2 * ROW];
  float* bufs[2] = { buf0, buf0 + ROW };

  const int i0 = ((int)blockIdx.x * WAVES + wave) * KP;
  // one division per wave; everything else increments with carry
  int b = i0 / SM1;
  int s = i0 - b * SM1;
  int bn = b, sn = s;  // tracker for the prefetched (next) pair

  float acc = 0.0f;

  // Prologue: kick off DMA for first row (batch row s+1 of pair i0)
  issue_row(batch + ((size_t)b * NS + (s + 1)) * ROW, bufs[0], lane);

#pragma unroll
  for (int k = 0; k < KP; ++k) {
    if (k + 1 < KP) {
      // Prefetch next row into the other buffer, then wait for current row:
      // ASYNCcnt completes in order, so <=8 means the first 8 (current) landed.
      advance(bn, sn);
      issue_row(batch + ((size_t)bn * NS + (sn + 1)) * ROW, bufs[(k + 1) & 1], lane);
      asm volatile("s_wait_asynccnt 8" ::: "memory");
    } else {
      asm volatile("s_wait_asynccnt 0" ::: "memory");
    }

    // Scan the 4 KB row in LDS for the single hot (==1.0) entry.
    const float* lrow = bufs[k & 1] + lane * 4;
    int hot = -1;
#pragma unroll
    for (int j = 0; j < 8; ++j) {
      float4 v = *(const float4*)(lrow + j * 128);
      const int c0 = j * 128 + lane * 4;
      if (v.x > 0.5f) hot = c0 + 0;
      if (v.y > 0.5f) hot = c0 + 1;
      if (v.z > 0.5f) hot = c0 + 2;
      if (v.w > 0.5f) hot = c0 + 3;
    }
    // wave32 butterfly max-reduce
#pragma unroll
    for (int off = 16; off >= 1; off >>= 1)
      hot = max(hot, __shfl_xor(hot, off, 32));

    if (lane == 0 && hot >= 0) {
      // one-hot dot == single gather from pred (bit-exact vs. the einsum)
      const float p = pred[((size_t)b * NS + s) * NQ + (hot & (NQ - 1))];
      if (p > 0.0f) {
        // a = 1 iff hot index is in the "correct" half (< Q)
        acc += (hot < NQ) ? logf(p) : logf(1.0f - p);
      }
    }
    advance(b, s);
  }

  // Block reduction: reuse smem after all waves are done with their buffers.
  __syncthreads();
  if (lane == 0) smem[wave] = acc;
  __syncthreads();
  if (tid == 0) {
    float t = 0.0f;
#pragma unroll
    for (int w = 0; w < WAVES; ++w) t += smem[w];
    partials[blockIdx.x] = t;
  }
}

__global__ void __launch_bounds__(256) dkt_loss_reduce(
    const float* __restrict__ partials, int n, float* __restrict__ out) {
  __shared__ float sm[8];
  float acc = 0.0f;
  for (int i = threadIdx.x; i < n; i += 256) acc += partials[i];
#pragma unroll
  for (int off = 16; off >= 1; off >>= 1) acc += __shfl_xor(acc, off, 32);
  const int lane = threadIdx.x & 31, wave = threadIdx.x >> 5;
  if (lane == 0) sm[wave] = acc;
  __syncthreads();
  if (threadIdx.x == 0) {
    float t = 0.0f;
#pragma unroll
    for (int w = 0; w < 8; ++w) t += sm[w];
    out[0] = -t;  // loss = -sum(ll)
  }
}

extern "C" void kernel_launch(void* const* d_in, const int* in_sizes, int n_in,
                              void* d_out, int out_size, void* d_ws, size_t ws_size,
                              hipStream_t stream) {
  const float* pred  = (const float*)d_in[0];   // [1024, 200, 512] f32
  const float* batch = (const float*)d_in[1];   // [1024, 200, 1024] f32
  float* partials = (float*)d_ws;

  const int npairs  = NB * SM1;                 // 203776
  const int nblocks = npairs / (WAVES * KP);    // 3184 (exact)

  dkt_loss_stage1<<<nblocks, BDIM, 0, stream>>>(pred, batch, partials);
  dkt_loss_reduce<<<1, 256, 0, stream>>>(partials, nblocks, (float*)d_out);
}